// SimpleAddEmbed_87823491269193
// MI455X (gfx1250) — compile-verified
//
#include <hip/hip_runtime.h>

typedef __attribute__((ext_vector_type(2))) float v2f;
typedef __attribute__((ext_vector_type(8))) float v8f;

#define VOCAB 50000
#define DIM   256
#define NTILE (VOCAB / 16)     // 3125 exactly
#define NTOK  (4 * 64 * 64 * 32)  // 524288 tokens
#define NCELL (4 * 64 * 64)       // 16384 cells

// Phase 1: p[v] = dot(E[v,:], W)  via V_WMMA_F32_16X16X4_F32.
// One wave per 16-row tile of E. A-matrix per ISA layout for 32-bit 16x4:
//   lane L (m=L&15, half=L>>4): VGPR0/1 hold K = 2*half, 2*half+1.
// B-matrix: W chunk replicated across all 16 columns -> every column of D
// holds the GEMV result; we read column 0 (lane 0 -> M=0..7, lane 16 -> M=8..15).
__global__ __launch_bounds__(128) void vocab_dot_wmma(const float* __restrict__ E,
                                                      const float* __restrict__ W,
                                                      float* __restrict__ p) {
  const int lane = threadIdx.x & 31;
  const int wave = threadIdx.x >> 5;
  const int tile = blockIdx.x * 4 + wave;
  if (tile >= NTILE) return;               // wave-uniform: EXEC stays all-1s for WMMA

  const int m    = lane & 15;
  const int half = lane >> 4;
  const float* erow = E + (size_t)(tile * 16 + m) * DIM + 2 * half;
  const float* wsub = W + 2 * half;

  v8f acc = {};
#pragma unroll 8
  for (int k = 0; k < DIM; k += 4) {
    v2f a  = *(const v2f*)(erow + k);      // {E[m, k+2h], E[m, k+2h+1]}
    v2f bw = *(const v2f*)(wsub + k);      // {W[k+2h],    W[k+2h+1]} replicated over N
    acc = __builtin_amdgcn_wmma_f32_16x16x4_f32(
        /*neg_a=*/false, a, /*neg_b=*/false, bw,
        /*c_mod=*/(short)0, acc, /*reuse_a=*/false, /*reuse_b=*/false);
  }

  if (m == 0) {                            // lanes 0 and 16 hold column N=0
    float* dst = p + tile * 16 + half * 8;
#pragma unroll
    for (int i = 0; i < 8; ++i) dst[i] = acc[i];
  }
}

// Phase 2: one thread per token; each wave32 covers exactly one cell (T=32).
// Coalesced index loads, scalar gathers from the 200KB L2-resident p table,
// butterfly reduce across the wave, lane 0 writes score + bias.
__global__ __launch_bounds__(256) void gather_reduce(const int* __restrict__ x,
                                                     const float* __restrict__ p,
                                                     const float* __restrict__ bias,
                                                     float* __restrict__ out) {
  const int tid  = blockIdx.x * 256 + threadIdx.x;   // token index, grid is exact
  const int lane = threadIdx.x & 31;

  float v = p[x[tid]];
#pragma unroll
  for (int off = 16; off > 0; off >>= 1)
    v += __shfl_xor(v, off, 32);

  if (lane == 0) out[tid >> 5] = v + bias[0];
}

extern "C" void kernel_launch(void* const* d_in, const int* in_sizes, int n_in,
                              void* d_out, int out_size, void* d_ws, size_t ws_size,
                              hipStream_t stream) {
  const int*   x = (const int*)d_in[0];    // [4,64,64,32] int32
  const float* E = (const float*)d_in[1];  // [50000,256] f32
  const float* W = (const float*)d_in[2];  // [256] f32
  const float* b = (const float*)d_in[3];  // [1] f32
  float* out = (float*)d_out;              // [4,64,64] f32
  float* p   = (float*)d_ws;               // 50000 f32 scratch (200 KB)

  // 3125 tiles, 4 waves per 128-thread block -> 782 blocks
  vocab_dot_wmma<<<(NTILE + 3) / 4, 128, 0, stream>>>(E, W, p);
  // 524288 tokens / 256 = 2048 blocks
  gather_reduce<<<NTOK / 256, 256, 0, stream>>>(x, p, b, out);
}